// AVID_CMA_59072980189422
// MI455X (gfx1250) — compile-verified
//
#include <hip/hip_runtime.h>

// ---------------------------------------------------------------------------
// AVID-CMA scoring on MI455X (gfx1250).
// Memory-bound gather + batched dot products. Uses V_WMMA_F32_16X16X4_F32
// (exact f32 matrix pipe) with A = {v[b], a[b]} contexts, B = 16 gathered
// memory rows, so every gathered row feeds two outputs in one pass.
// ---------------------------------------------------------------------------

#define DIM   128
#define BS    512
#define NNEG  1024
#define POSK  32

// Output offsets (flat, reference return order)
#define OFF_IV2AP 0                         // inst_v2a_pos  [512]
#define OFF_IV2AN 512                       // inst_v2a_neg  [512*1024]
#define OFF_IA2VP (512 + 524288)            // inst_a2v_pos  [512]
#define OFF_IA2VN (OFF_IA2VP + 512)         // inst_a2v_neg  [512*1024]
#define OFF_PV2VP (OFF_IA2VN + 524288)      // pos_v2v_pos   [512*32]
#define OFF_PV2VN (OFF_PV2VP + 16384)       // pos_v2v_neg   [512*1024]
#define OFF_PA2AP (OFF_PV2VN + 524288)      // pos_a2a_pos   [512*32]
#define OFF_PA2AN (OFF_PA2AP + 16384)       // pos_a2a_neg   [512*1024]

typedef __attribute__((ext_vector_type(2))) float v2f;
typedef __attribute__((ext_vector_type(8))) float v8f;

// ---------------------------------------------------------------------------
// Kernel 1: L2-normalize video/audio embeddings into workspace.
// One wave per row; 4 floats/lane; wave32 xor-shuffle reduction.
// ---------------------------------------------------------------------------
__global__ __launch_bounds__(256) void avid_norm_kernel(
    const float* __restrict__ video, const float* __restrict__ audio,
    float* __restrict__ vnorm, float* __restrict__ anorm)
{
    const int gw   = blockIdx.x * 8 + (threadIdx.x >> 5);   // 0..1023
    const int lane = threadIdx.x & 31;
    const int row  = gw & (BS - 1);
    const float* src = (gw < BS) ? video : audio;
    float*       dst = (gw < BS) ? vnorm : anorm;

    const float4 x = *(const float4*)(src + row * DIM + 4 * lane);
    float ss = x.x * x.x + x.y * x.y + x.z * x.z + x.w * x.w;
    #pragma unroll
    for (int m = 16; m >= 1; m >>= 1) ss += __shfl_xor(ss, m, 32);
    const float s = 1.0f / fmaxf(sqrtf(ss), 1e-12f);
    float4 yv; yv.x = x.x * s; yv.y = x.y * s; yv.z = x.z * s; yv.w = x.w * s;
    *(float4*)(dst + row * DIM + 4 * lane) = yv;
}

// ---------------------------------------------------------------------------
// Kernel 2: negatives. One wave per (b, 16-negative tile).
//   A (16x4 f32): row0 = v[b], row1 = a[b], rows 2..15 duplicate v[b] (unused C rows).
//   B (4x16 f32): 16 gathered rows (view1 stream -> C1, view2 stream -> C2).
//   C1(0,n)=v.v_neg -> pos_v2v_neg   C1(1,n)=a.v_neg -> inst_a2v_neg
//   C2(0,n)=v.a_neg -> inst_v2a_neg  C2(1,n)=a.a_neg -> pos_a2a_neg
// Per-lane WMMA layout (ISA 7.12.2): lane l holds row (l&15), K pair at
// k0 + 2*(l>>4) for both A and B -> plain float2 loads.
// ---------------------------------------------------------------------------
__global__ __launch_bounds__(256) void avid_neg_kernel(
    const float* __restrict__ view1, const float* __restrict__ view2,
    const int* __restrict__ neg_idx,
    const float* __restrict__ vnorm, const float* __restrict__ anorm,
    float* __restrict__ out)
{
    const int b    = blockIdx.x;
    const int wave = threadIdx.x >> 5;
    const int lane = threadIdx.x & 31;
    const int nt   = blockIdx.y * 8 + wave;      // 0..63
    const int lm   = lane & 15;
    const int half = lane >> 4;

    const int idx = neg_idx[b * NNEG + nt * 16 + lm];
    const float* r1  = view1 + (size_t)idx * DIM;
    const float* r2  = view2 + (size_t)idx * DIM;
    const float* ctx = (lm == 1) ? (anorm + b * DIM) : (vnorm + b * DIM);

    v8f c1 = {};
    v8f c2 = {};
    #pragma unroll
    for (int k0 = 0; k0 < DIM; k0 += 4) {
        const int kk = k0 + 2 * half;
        const v2f af = *(const v2f*)(ctx + kk);
        const v2f b1 = *(const v2f*)(r1 + kk);
        const v2f b2 = *(const v2f*)(r2 + kk);
        c1 = __builtin_amdgcn_wmma_f32_16x16x4_f32(false, af, false, b1,
                                                   (short)0, c1, false, false);
        c2 = __builtin_amdgcn_wmma_f32_16x16x4_f32(false, af, false, b2,
                                                   (short)0, c2, false, false);
    }

    if (lane < 16) {
        const int o = b * NNEG + nt * 16 + lane;   // lanes 0..15 carry N=0..15
        out[OFF_PV2VN + o] = c1[0] / 0.07f;
        out[OFF_IA2VN + o] = c1[1] / 0.07f;
        out[OFF_IV2AN + o] = c2[0] / 0.07f;
        out[OFF_PA2AN + o] = c2[1] / 0.07f;
    }
}

// ---------------------------------------------------------------------------
// Kernel 3: positives. Same tile shape, 2 tiles of 16 per batch item.
//   C1(0,n)=v.v_pos -> pos_v2v_pos ; C2(1,n)=a.a_pos -> pos_a2a_pos
// ---------------------------------------------------------------------------
__global__ __launch_bounds__(64) void avid_pos_kernel(
    const float* __restrict__ view1, const float* __restrict__ view2,
    const int* __restrict__ pos_idx,
    const float* __restrict__ vnorm, const float* __restrict__ anorm,
    float* __restrict__ out)
{
    const int b    = blockIdx.x;
    const int wave = threadIdx.x >> 5;           // 0..1
    const int lane = threadIdx.x & 31;
    const int lm   = lane & 15;
    const int half = lane >> 4;

    const int idx = pos_idx[b * POSK + wave * 16 + lm];
    const float* r1  = view1 + (size_t)idx * DIM;
    const float* r2  = view2 + (size_t)idx * DIM;
    const float* ctx = (lm == 1) ? (anorm + b * DIM) : (vnorm + b * DIM);

    v8f c1 = {};
    v8f c2 = {};
    #pragma unroll
    for (int k0 = 0; k0 < DIM; k0 += 4) {
        const int kk = k0 + 2 * half;
        const v2f af = *(const v2f*)(ctx + kk);
        const v2f b1 = *(const v2f*)(r1 + kk);
        const v2f b2 = *(const v2f*)(r2 + kk);
        c1 = __builtin_amdgcn_wmma_f32_16x16x4_f32(false, af, false, b1,
                                                   (short)0, c1, false, false);
        c2 = __builtin_amdgcn_wmma_f32_16x16x4_f32(false, af, false, b2,
                                                   (short)0, c2, false, false);
    }

    if (lane < 16) {
        const int o = b * POSK + wave * 16 + lane;
        out[OFF_PV2VP + o] = c1[0] / 0.07f;
        out[OFF_PA2AP + o] = c2[1] / 0.07f;
    }
}

// ---------------------------------------------------------------------------
// Kernel 4: self scores. One wave per batch item; two 128-dim dots.
// ---------------------------------------------------------------------------
__global__ __launch_bounds__(256) void avid_self_kernel(
    const float* __restrict__ view1, const float* __restrict__ view2,
    const int* __restrict__ y,
    const float* __restrict__ vnorm, const float* __restrict__ anorm,
    float* __restrict__ out)
{
    const int b    = blockIdx.x * 8 + (threadIdx.x >> 5);   // 0..511
    const int lane = threadIdx.x & 31;
    const int yi   = y[b];

    const float4 vn = *(const float4*)(vnorm + b * DIM + 4 * lane);
    const float4 an = *(const float4*)(anorm + b * DIM + 4 * lane);
    const float4 m1 = *(const float4*)(view1 + (size_t)yi * DIM + 4 * lane);
    const float4 m2 = *(const float4*)(view2 + (size_t)yi * DIM + 4 * lane);

    float dva = vn.x * m2.x + vn.y * m2.y + vn.z * m2.z + vn.w * m2.w;
    float dav = an.x * m1.x + an.y * m1.y + an.z * m1.z + an.w * m1.w;
    #pragma unroll
    for (int m = 16; m >= 1; m >>= 1) {
        dva += __shfl_xor(dva, m, 32);
        dav += __shfl_xor(dav, m, 32);
    }
    if (lane == 0) {
        out[OFF_IV2AP + b] = dva / 0.07f;   // v . view2_mem[y]
        out[OFF_IA2VP + b] = dav / 0.07f;   // a . view1_mem[y]
    }
}

// ---------------------------------------------------------------------------
extern "C" void kernel_launch(void* const* d_in, const int* in_sizes, int n_in,
                              void* d_out, int out_size, void* d_ws, size_t ws_size,
                              hipStream_t stream) {
    const float* video  = (const float*)d_in[0];   // [512,128]
    const float* audio  = (const float*)d_in[1];   // [512,128]
    const float* view1  = (const float*)d_in[2];   // [500000,128]
    const float* view2  = (const float*)d_in[3];   // [500000,128]
    const int*   y      = (const int*)  d_in[4];   // [512]
    const int*   posidx = (const int*)  d_in[5];   // [512,32]
    const int*   negidx = (const int*)  d_in[6];   // [512,1024]
    float* out = (float*)d_out;

    float* vnorm = (float*)d_ws;                   // [512,128]
    float* anorm = vnorm + BS * DIM;               // [512,128]
    (void)in_sizes; (void)n_in; (void)out_size; (void)ws_size;

    // 1) normalize embeddings (1024 waves)
    avid_norm_kernel<<<128, 256, 0, stream>>>(video, audio, vnorm, anorm);
    // 2) negatives: 512 x 64 tiles, 8 waves/block
    avid_neg_kernel<<<dim3(BS, 8), 256, 0, stream>>>(view1, view2, negidx,
                                                     vnorm, anorm, out);
    // 3) positives: 512 x 2 tiles
    avid_pos_kernel<<<BS, 64, 0, stream>>>(view1, view2, posidx,
                                           vnorm, anorm, out);
    // 4) self scores: 512 waves
    avid_self_kernel<<<64, 256, 0, stream>>>(view1, view2, y,
                                             vnorm, anorm, out);
}